// _GGUFQuantizedBase_84043920048498
// MI455X (gfx1250) — compile-verified
//
#include <hip/hip_runtime.h>
#include <stdint.h>

// GGUF Q8_0 dequant for MI455X (gfx1250, wave32).
// Memory-bound: 256MiB q(int32) + 8MiB scales in, 256MiB f32 out (~22us @ 23.3TB/s).
// Uses CDNA5 async global->LDS loads (ASYNCcnt) as a zero-VGPR-cost prefetch
// pipeline; each wave stages 512B per async op and consumes only its own LDS
// region, so the only sync needed is the wave-private s_wait_asynccnt.

#define THREADS 256
#define NBLOCKS 2048
#define ITERS   32
// NBLOCKS * THREADS * ITERS = 16,777,216 chunks of 4 quants = 67,108,864 quants.

__device__ __forceinline__ void async_ld_b128(uint32_t lds_addr, uint32_t voff,
                                              const void* sbase) {
  // GVS form: mem_addr = SGPR64 + sext(VGPR32). Offsets max 256MiB, fits i32.
  asm volatile("global_load_async_to_lds_b128 %0, %1, %2"
               :: "v"(lds_addr), "v"(voff), "s"(sbase)
               : "memory");
}

#define WAIT_ASYNCCNT(N) asm volatile("s_wait_asynccnt " #N ::: "memory")

__global__ __launch_bounds__(THREADS, 1)
void q8_0_dequant_kernel(const int* __restrict__ q,
                         const float* __restrict__ scales,
                         float* __restrict__ out)
{
  // Two staging buffers: 256 threads x 16B each = 4KiB per buffer.
  __shared__ __align__(16) int lds[2][THREADS * 4];

  const uint32_t tid    = threadIdx.x;
  const uint32_t chunk0 = blockIdx.x * (THREADS * ITERS) + tid;

  // Low 32 bits of a generic pointer to __shared__ = LDS byte offset.
  const uint32_t ldsA = (uint32_t)(uintptr_t)(&lds[0][tid * 4]);
  const uint32_t ldsB = (uint32_t)(uintptr_t)(&lds[1][tid * 4]);

  // Prologue: stage iteration 0 into buffer A.
  async_ld_b128(ldsA, chunk0 * 16u, (const void*)q);

  uint32_t chunk = chunk0;
  for (int it = 0; it < ITERS; ++it) {
    const bool even = (it & 1) == 0;

    if (it + 1 < ITERS) {
      // Prefetch next chunk into the other buffer, then wait until only that
      // one async op is outstanding (async loads complete in order).
      async_ld_b128(even ? ldsB : ldsA, (chunk + THREADS) * 16u, (const void*)q);
      WAIT_ASYNCCNT(0x1);
    } else {
      WAIT_ASYNCCNT(0x0);
    }

    // Read back own 16B (ds_load_b128); wave-private, no barrier required.
    const int* src = even ? &lds[0][tid * 4] : &lds[1][tid * 4];
    const int4 qv  = *reinterpret_cast<const int4*>(src);

    // 8 chunks (of 4 quants) per Q8_0 block of 32 -> block index = chunk >> 3.
    const _Float16 d = (_Float16)scales[chunk >> 3];

    // Match reference rounding: fp16(q-128) * fp16(scale), widened to f32.
    float4 o;
    o.x = (float)(d * (_Float16)(qv.x - 128));
    o.y = (float)(d * (_Float16)(qv.y - 128));
    o.z = (float)(d * (_Float16)(qv.z - 128));
    o.w = (float)(d * (_Float16)(qv.w - 128));

    reinterpret_cast<float4*>(out)[chunk] = o;

    chunk += THREADS;
  }
}

extern "C" void kernel_launch(void* const* d_in, const int* in_sizes, int n_in,
                              void* d_out, int out_size, void* d_ws, size_t ws_size,
                              hipStream_t stream) {
  const int*   q      = (const int*)d_in[0];    // [N_BLOCKS, 32] int32 (0..255)
  const float* scales = (const float*)d_in[1];  // [N_BLOCKS] f32
  float*       out    = (float*)d_out;          // [8192*8192] (fp16 values, f32 carrier)
  (void)in_sizes; (void)n_in; (void)out_size; (void)d_ws; (void)ws_size;

  q8_0_dequant_kernel<<<NBLOCKS, THREADS, 0, stream>>>(q, scales, out);
}